// GraphRerankerGNN_21251498180624
// MI455X (gfx1250) — compile-verified
//
#include <hip/hip_runtime.h>
#include <hip/hip_bf16.h>
#include <math.h>

typedef __attribute__((ext_vector_type(2))) float v2f;
typedef __attribute__((ext_vector_type(8))) float v8f;

#define HID 128

// ---------------------------------------------------------------- utilities
__global__ void zero_f32(float* __restrict__ p, long long n) {
  long long i = (long long)blockIdx.x * blockDim.x + threadIdx.x;
  long long stride = (long long)gridDim.x * blockDim.x;
  for (; i < n; i += stride) p[i] = 0.0f;
}

__global__ void deg_kernel(const int* __restrict__ dst, float* __restrict__ deg, int E) {
  int e = blockIdx.x * blockDim.x + threadIdx.x;
  if (e < E) atomicAdd(&deg[dst[e]], 1.0f);
}

__global__ void invdeg_kernel(float* __restrict__ deg, int N) {
  int i = blockIdx.x * blockDim.x + threadIdx.x;
  if (i < N) deg[i] = 1.0f / (deg[i] + 1.0f);
}

// ------------------------------------------------- edge gather/scatter (agg += h[src])
// 32 lanes per edge, float4 per lane: 512B coalesced gather, 4 f32 atomics per lane.
__global__ void scatter_kernel(const float* __restrict__ h,
                               const int* __restrict__ src,
                               const int* __restrict__ dst,
                               float* __restrict__ agg, int E) {
  long long tid = (long long)blockIdx.x * blockDim.x + threadIdx.x;
  int e = (int)(tid >> 5);
  if (e >= E) return;
  int c = ((int)tid & 31) * 4;
  int s = src[e], d = dst[e];
  float4 v = *(const float4*)(h + (long long)s * HID + c);
  float* ap = agg + (long long)d * HID + c;
  atomicAdd(ap + 0, v.x);
  atomicAdd(ap + 1, v.y);
  atomicAdd(ap + 2, v.z);
  atomicAdd(ap + 3, v.w);
}

// ------------------------------------------------- m = (h + agg) * inv_deg[node]
__global__ void prep_m_kernel(const float* __restrict__ h,
                              const float* __restrict__ agg,
                              const float* __restrict__ inv,
                              float* __restrict__ m, int N) {
  long long tid = (long long)blockIdx.x * blockDim.x + threadIdx.x;
  int n = (int)(tid >> 5);
  if (n >= N) return;
  int c = ((int)tid & 31) * 4;
  float iv = inv[n];
  long long base = (long long)n * HID + c;
  float4 a = *(const float4*)(h + base);
  float4 g = *(const float4*)(agg + base);
  float4 r;
  r.x = (a.x + g.x) * iv;
  r.y = (a.y + g.y) * iv;
  r.z = (a.z + g.z) * iv;
  r.w = (a.w + g.w) * iv;
  *(float4*)(m + base) = r;
}

// ------------------------------------------------- fp32 WMMA GEMM: D = relu(A @ W + b)
// A: [N x 128] row-major, W: [128 x OUT] row-major, D: [N x OUT].
// One wave -> one 16x16 tile; block = OUT/16 waves sharing a 16-row A tile in LDS.
// V_WMMA_F32_16X16X4_F32 fragment layouts (ISA 7.12.2):
//   A 16x4 : lanes 0-15 rows M, VGPR{0,1} = K{0,1}; lanes 16-31 = K{2,3}
//   B 4x16 : lanes 0-15 cols N, VGPR{0,1} = K{0,1}; lanes 16-31 = K{2,3}
//   C/D    : VGPR i, lane L -> row = i + (L<16?0:8), col = L&15
template <int OUT>
__global__ __launch_bounds__(OUT * 2) void gemm_relu_wmma(
    const float* __restrict__ A, const float* __restrict__ W,
    const float* __restrict__ bias, float* __restrict__ D, int N) {
  constexpr int PAD = 132;  // kill 16-way bank conflict on column reads
  __shared__ float As[16 * PAD];

  const int row0 = blockIdx.x * 16;
  const int t = threadIdx.x;
  constexpr int NT = OUT * 2;

  // stage 16x128 A tile into LDS (coalesced float4 global loads)
  for (int idx = t * 4; idx < 16 * HID; idx += NT * 4) {
    int r = idx >> 7;
    int k = idx & 127;
    float4 v = *(const float4*)(A + (long long)(row0 + r) * HID + k);
    As[r * PAD + k + 0] = v.x;
    As[r * PAD + k + 1] = v.y;
    As[r * PAD + k + 2] = v.z;
    As[r * PAD + k + 3] = v.w;
  }
  __syncthreads();

  const int lane = t & 31;
  const int wave = t >> 5;
  const int col0 = wave * 16;
  const int r = lane & 15;            // row (A) / col (B,C) within tile
  const int ksel = (lane >> 4) << 1;  // 0 for lanes 0-15, 2 for lanes 16-31

  float bv = bias[col0 + r];
  v8f acc = {bv, bv, bv, bv, bv, bv, bv, bv};

  const float* wp = W + col0 + r;
#pragma unroll 4
  for (int kk = 0; kk < HID; kk += 4) {
    v2f a = *(const v2f*)&As[r * PAD + kk + ksel];
    v2f b;
    b.x = wp[(kk + ksel) * OUT];
    b.y = wp[(kk + ksel + 1) * OUT];
    acc = __builtin_amdgcn_wmma_f32_16x16x4_f32(
        false, a, false, b, (short)0, acc, false, false);
  }

  const int col = col0 + r;
  const int rbase = row0 + ((lane >> 4) << 3);
#pragma unroll
  for (int i = 0; i < 8; i++) {
    float v = acc[i];
    v = v > 0.0f ? v : 0.0f;
    D[(long long)(rbase + i) * OUT + col] = v;
  }
}

// ------------------------------------------------- head stage 2 + blend
__global__ void head2_kernel(const float* __restrict__ hd,   // [N x 64]
                             const float* __restrict__ Wh2,  // [64]
                             const float* __restrict__ bh2,  // [1]
                             const float* __restrict__ orig,
                             const float* __restrict__ alpha_logit,
                             float* __restrict__ out, int N) {
  int n = blockIdx.x * blockDim.x + threadIdx.x;
  if (n >= N) return;
  float acc = bh2[0];
  const float* hp = hd + (long long)n * 64;
#pragma unroll 8
  for (int j = 0; j < 64; j++) acc += hp[j] * Wh2[j];
  float al = alpha_logit[0];
  float alpha = 1.0f / (1.0f + __expf(-al));
  out[n] = alpha * orig[n] + (1.0f - alpha) * acc;
}

// ---------------------------------------------------------------- launch
extern "C" void kernel_launch(void* const* d_in, const int* in_sizes, int n_in,
                              void* d_out, int out_size, void* d_ws, size_t ws_size,
                              hipStream_t stream) {
  const float* x    = (const float*)d_in[0];
  const int*   ei   = (const int*)d_in[1];
  const float* orig = (const float*)d_in[2];
  const float* W1   = (const float*)d_in[3];
  const float* b1   = (const float*)d_in[4];
  const float* W2   = (const float*)d_in[5];
  const float* b2   = (const float*)d_in[6];
  const float* Wh1  = (const float*)d_in[7];
  const float* bh1  = (const float*)d_in[8];
  const float* Wh2  = (const float*)d_in[9];
  const float* bh2  = (const float*)d_in[10];
  const float* alog = (const float*)d_in[11];
  float* out = (float*)d_out;

  const int N = in_sizes[0] / HID;  // 50000
  const int E = in_sizes[1] / 2;    // 800000
  const int* src = ei;
  const int* dst = ei + E;

  // workspace layout: inv[N] | agg[N*128] | m[N*128] | h[N*128]
  float* inv = (float*)d_ws;
  float* agg = inv + N;
  float* mbf = agg + (size_t)N * HID;
  float* hbf = mbf + (size_t)N * HID;

  const int edgeBlocks    = (E + 255) / 256;
  const int nodeBlocks    = (N + 255) / 256;
  const int scatterBlocks = (int)(((long long)E * 32 + 255) / 256);
  const int prepBlocks    = (int)(((long long)N * 32 + 255) / 256);
  const int tileBlocks    = (N + 15) / 16;  // 3125

  // degree -> inv_deg (zero inv+agg in one pass: contiguous)
  zero_f32<<<2048, 256, 0, stream>>>(inv, (long long)N * (1 + HID));
  deg_kernel<<<edgeBlocks, 256, 0, stream>>>(dst, inv, E);
  invdeg_kernel<<<nodeBlocks, 256, 0, stream>>>(inv, N);

  // layer 1
  scatter_kernel<<<scatterBlocks, 256, 0, stream>>>(x, src, dst, agg, E);
  prep_m_kernel<<<prepBlocks, 256, 0, stream>>>(x, agg, inv, mbf, N);
  gemm_relu_wmma<HID><<<tileBlocks, 256, 0, stream>>>(mbf, W1, b1, hbf, N);

  // layer 2
  zero_f32<<<2048, 256, 0, stream>>>(agg, (long long)N * HID);
  scatter_kernel<<<scatterBlocks, 256, 0, stream>>>(hbf, src, dst, agg, E);
  prep_m_kernel<<<prepBlocks, 256, 0, stream>>>(hbf, agg, inv, mbf, N);
  gemm_relu_wmma<HID><<<tileBlocks, 256, 0, stream>>>(mbf, W2, b2, agg, N);  // h2 -> agg

  // head: relu(h2 @ Wh1 + bh1) -> mbf[N x 64], then blend
  gemm_relu_wmma<64><<<tileBlocks, 128, 0, stream>>>(agg, Wh1, bh1, mbf, N);
  head2_kernel<<<nodeBlocks, 256, 0, stream>>>(mbf, Wh2, bh2, orig, alog, out, N);
}